// AdvancedStockGNN_43903155699776
// MI455X (gfx1250) — compile-verified
//
#include <hip/hip_runtime.h>
#include <float.h>

// GAT 2-layer forward for MI455X (gfx1250, wave32).
// GEMMs use V_WMMA_F32_16X16X4_F32 (exact f32 matrix path).
// Edge softmax+aggregation is L2-resident atomic-bound; h1/out1 (51MB each)
// fit in the 192MB L2 so gathers/atomics never touch HBM steady-state.

typedef float v2f __attribute__((ext_vector_type(2)));
typedef float v8f __attribute__((ext_vector_type(8)));

#define NEG_SLOPE 0.2f
#define EPSV 1e-16f

__device__ __forceinline__ float lrelu(float x) { return x > 0.f ? x : NEG_SLOPE * x; }

// float atomic max via signed-int max (v>=0) / unsigned-int min (v<0).
__device__ __forceinline__ void atomicMaxF(float* addr, float v) {
  if (v >= 0.f) atomicMax((int*)addr, __float_as_int(v));
  else          atomicMin((unsigned int*)addr, __float_as_uint(v));
}

__device__ __forceinline__ void edge_sd(const int* __restrict__ ei, int e, int E,
                                        int& s, int& d) {
  if (e < E) { s = ei[e]; d = ei[E + e]; }
  else       { s = d = e - E; }   // self loops appended after real edges
}

// ---------------- init: zero accumulators, -inf maxes ----------------
__global__ void k_init(float* o1, float* s1, float* m1, float* outF,
                       float* s2, float* m2, int N) {
  int i = blockIdx.x * blockDim.x + threadIdx.x;
  if (i < N * 256) o1[i] = 0.f;
  if (i < N * 16)  outF[i] = 0.f;
  if (i < N * 8)   { s1[i] = 0.f; m1[i] = -FLT_MAX; }
  if (i < N)       { s2[i] = 0.f; m2[i] = -FLT_MAX; }
}

// ---------------- GEMM1: H[N,256] = X[N,128] @ W[128,256] ----------------
// One wave per 16x16 tile. A: row=lane&15, K=2*(lane>>4)+{0,1} (contiguous f2).
// B: col=lane&15, same K striping. D: row=vgpr+8*(lane>>4), col=lane&15.
__global__ void k_gemm1(const float* __restrict__ X, const float* __restrict__ W,
                        float* __restrict__ H, int MT) {
  int gid  = blockIdx.x * blockDim.x + threadIdx.x;
  int wave = gid >> 5;
  int lane = gid & 31;
  if (wave >= MT * 16) return;
  int mt = wave >> 4, nt = wave & 15;
  int l15 = lane & 15, kg = lane >> 4;
  int row = mt * 16 + l15;
  int col = nt * 16 + l15;
  v8f acc = {};
  const float* xr = X + (size_t)row * 128 + 2 * kg;
#pragma unroll 4
  for (int k = 0; k < 128; k += 4) {
    v2f a = *(const v2f*)(xr + k);
    v2f b;
    b.x = W[(size_t)(k + 2 * kg + 0) * 256 + col];
    b.y = W[(size_t)(k + 2 * kg + 1) * 256 + col];
    acc = __builtin_amdgcn_wmma_f32_16x16x4_f32(false, a, false, b,
                                                (short)0, acc, false, false);
  }
  float* hp = H + (size_t)(mt * 16 + 8 * kg) * 256 + col;
#pragma unroll
  for (int v = 0; v < 8; ++v) hp[(size_t)v * 256] = acc[v];
}

// ---------------- GEMM2: H2[N,16] = A[N,256] @ W[256,16] ----------------
__global__ void k_gemm2(const float* __restrict__ A, const float* __restrict__ W,
                        float* __restrict__ H2, int MT) {
  int gid  = blockIdx.x * blockDim.x + threadIdx.x;
  int wave = gid >> 5;
  int lane = gid & 31;
  if (wave >= MT) return;
  int l15 = lane & 15, kg = lane >> 4;
  int row = wave * 16 + l15;
  v8f acc = {};
  const float* ar = A + (size_t)row * 256 + 2 * kg;
#pragma unroll 4
  for (int k = 0; k < 256; k += 4) {
    v2f a = *(const v2f*)(ar + k);
    v2f b;
    b.x = W[(size_t)(k + 2 * kg + 0) * 16 + l15];
    b.y = W[(size_t)(k + 2 * kg + 1) * 16 + l15];
    acc = __builtin_amdgcn_wmma_f32_16x16x4_f32(false, a, false, b,
                                                (short)0, acc, false, false);
  }
  float* hp = H2 + (size_t)(wave * 16 + 8 * kg) * 16 + l15;
#pragma unroll
  for (int v = 0; v < 8; ++v) hp[(size_t)v * 16] = acc[v];
}

// ---------------- attention coefficients, layer 1 (thread per node,head) ----
__global__ void k_att1(const float* __restrict__ H, const float* __restrict__ ATS,
                       const float* __restrict__ ATD, float* __restrict__ as_,
                       float* __restrict__ ad_, int N) {
  int i = blockIdx.x * blockDim.x + threadIdx.x;
  if (i >= N * 8) return;
  int n = i >> 3, h = i & 7;
  const float4* hp = (const float4*)(H + (size_t)n * 256 + h * 32);
  const float4* sp = (const float4*)(ATS + h * 32);
  const float4* dp = (const float4*)(ATD + h * 32);
  float s = 0.f, d = 0.f;
#pragma unroll
  for (int j = 0; j < 8; ++j) {
    float4 v = hp[j], a = sp[j], b = dp[j];
    s += v.x * a.x + v.y * a.y + v.z * a.z + v.w * a.w;
    d += v.x * b.x + v.y * b.y + v.z * b.z + v.w * b.w;
  }
  as_[i] = s; ad_[i] = d;
}

// ---------------- edge softmax passes, layer 1 ----------------
__global__ void k_emax1(const int* __restrict__ ei, const float* __restrict__ as_,
                        const float* __restrict__ ad_, float* m1, int E, int E2) {
  int i = blockIdx.x * blockDim.x + threadIdx.x;
  if (i >= E2 * 8) return;
  int e = i >> 3, h = i & 7;
  int s, d; edge_sd(ei, e, E, s, d);
  float ev = lrelu(as_[s * 8 + h] + ad_[d * 8 + h]);
  atomicMaxF(&m1[d * 8 + h], ev);
}

__global__ void k_esum1(const int* __restrict__ ei, const float* __restrict__ as_,
                        const float* __restrict__ ad_, const float* __restrict__ m1,
                        float* s1, int E, int E2) {
  int i = blockIdx.x * blockDim.x + threadIdx.x;
  if (i >= E2 * 8) return;
  int e = i >> 3, h = i & 7;
  int s, d; edge_sd(ei, e, E, s, d);
  float ev = lrelu(as_[s * 8 + h] + ad_[d * 8 + h]);
  atomicAdd(&s1[d * 8 + h], expf(ev - m1[d * 8 + h]));
}

// ---------------- aggregation, layer 1: 64 lanes/edge, float4 each ----------
__global__ void k_agg1(const int* __restrict__ ei, const float* __restrict__ H,
                       const float* __restrict__ as_, const float* __restrict__ ad_,
                       const float* __restrict__ m1, const float* __restrict__ s1,
                       float* out1, int E, int E2) {
  int i = blockIdx.x * blockDim.x + threadIdx.x;
  if (i >= E2 * 64) return;
  int e = i >> 6, j = i & 63, h = j >> 3;
  int s, d; edge_sd(ei, e, E, s, d);
  float ev = lrelu(as_[s * 8 + h] + ad_[d * 8 + h]);
  float alpha = expf(ev - m1[d * 8 + h]) / (s1[d * 8 + h] + EPSV);
  float4 hv = *(const float4*)(H + (size_t)s * 256 + j * 4);
  float* op = out1 + (size_t)d * 256 + j * 4;
  atomicAdd(op + 0, hv.x * alpha);
  atomicAdd(op + 1, hv.y * alpha);
  atomicAdd(op + 2, hv.z * alpha);
  atomicAdd(op + 3, hv.w * alpha);
}

// ---------------- bias + ELU (in place on out1) ----------------
__global__ void k_elubias(float* o1, const float* __restrict__ bias, int N) {
  int i = blockIdx.x * blockDim.x + threadIdx.x;
  if (i >= N * 256) return;
  float v = o1[i] + bias[i & 255];
  o1[i] = v > 0.f ? v : expm1f(v);
}

// ---------------- attention coefficients, layer 2 (thread per node) ---------
__global__ void k_att2(const float* __restrict__ H2, const float* __restrict__ ATS,
                       const float* __restrict__ ATD, float* as2, float* ad2, int N) {
  int n = blockIdx.x * blockDim.x + threadIdx.x;
  if (n >= N) return;
  float s = 0.f, d = 0.f;
#pragma unroll
  for (int c = 0; c < 16; ++c) {
    float v = H2[(size_t)n * 16 + c];
    s += v * ATS[c]; d += v * ATD[c];
  }
  as2[n] = s; ad2[n] = d;
}

__global__ void k_emax2(const int* __restrict__ ei, const float* __restrict__ as2,
                        const float* __restrict__ ad2, float* m2, int E, int E2) {
  int e = blockIdx.x * blockDim.x + threadIdx.x;
  if (e >= E2) return;
  int s, d; edge_sd(ei, e, E, s, d);
  atomicMaxF(&m2[d], lrelu(as2[s] + ad2[d]));
}

__global__ void k_esum2(const int* __restrict__ ei, const float* __restrict__ as2,
                        const float* __restrict__ ad2, const float* __restrict__ m2,
                        float* s2, int E, int E2) {
  int e = blockIdx.x * blockDim.x + threadIdx.x;
  if (e >= E2) return;
  int s, d; edge_sd(ei, e, E, s, d);
  float ev = lrelu(as2[s] + ad2[d]);
  atomicAdd(&s2[d], expf(ev - m2[d]));
}

// ---------------- aggregation, layer 2: 4 lanes/edge ----------------
__global__ void k_agg2(const int* __restrict__ ei, const float* __restrict__ H2,
                       const float* __restrict__ as2, const float* __restrict__ ad2,
                       const float* __restrict__ m2, const float* __restrict__ s2,
                       float* out, int E, int E2) {
  int i = blockIdx.x * blockDim.x + threadIdx.x;
  if (i >= E2 * 4) return;
  int e = i >> 2, p = i & 3;
  int s, d; edge_sd(ei, e, E, s, d);
  float ev = lrelu(as2[s] + ad2[d]);
  float alpha = expf(ev - m2[d]) / (s2[d] + EPSV);
  float4 hv = *(const float4*)(H2 + (size_t)s * 16 + p * 4);
  float* op = out + (size_t)d * 16 + p * 4;
  atomicAdd(op + 0, hv.x * alpha);
  atomicAdd(op + 1, hv.y * alpha);
  atomicAdd(op + 2, hv.z * alpha);
  atomicAdd(op + 3, hv.w * alpha);
}

__global__ void k_bias2(float* out, const float* __restrict__ bias, int N) {
  int i = blockIdx.x * blockDim.x + threadIdx.x;
  if (i >= N * 16) return;
  out[i] += bias[i & 15];
}

static inline int cdiv(long long a, int b) { return (int)((a + b - 1) / b); }

extern "C" void kernel_launch(void* const* d_in, const int* in_sizes, int n_in,
                              void* d_out, int out_size, void* d_ws, size_t ws_size,
                              hipStream_t stream) {
  const float* x    = (const float*)d_in[0];
  const int*   ei   = (const int*)d_in[1];     // [2,E] flattened: src row then dst row
  const float* W1   = (const float*)d_in[2];
  const float* ats1 = (const float*)d_in[3];
  const float* atd1 = (const float*)d_in[4];
  const float* b1   = (const float*)d_in[5];
  const float* W2   = (const float*)d_in[6];
  const float* ats2 = (const float*)d_in[7];
  const float* atd2 = (const float*)d_in[8];
  const float* b2   = (const float*)d_in[9];
  float* out = (float*)d_out;

  const int N  = in_sizes[0] / 128;   // 50000 (multiple of 16)
  const int E  = in_sizes[1] / 2;     // 800000
  const int E2 = E + N;               // + self loops
  const int MT = N / 16;

  // workspace layout (floats): 113 MB total
  float* ws  = (float*)d_ws;
  float* h1  = ws;                         // N*256
  float* o1  = h1  + (size_t)N * 256;      // N*256 (aggregated, then ELU'd)
  float* as1 = o1  + (size_t)N * 256;      // N*8
  float* ad1 = as1 + (size_t)N * 8;        // N*8
  float* m1  = ad1 + (size_t)N * 8;        // N*8
  float* s1  = m1  + (size_t)N * 8;        // N*8
  float* h2  = s1  + (size_t)N * 8;        // N*16
  float* as2 = h2  + (size_t)N * 16;       // N
  float* ad2 = as2 + (size_t)N;            // N
  float* m2  = ad2 + (size_t)N;            // N
  float* s2  = m2  + (size_t)N;            // N

  const int B = 256;
  k_init<<<cdiv((long long)N * 256, B), B, 0, stream>>>(o1, s1, m1, out, s2, m2, N);
  k_gemm1<<<cdiv((long long)MT * 16 * 32, B), B, 0, stream>>>(x, W1, h1, MT);
  k_att1<<<cdiv((long long)N * 8, B), B, 0, stream>>>(h1, ats1, atd1, as1, ad1, N);
  k_emax1<<<cdiv((long long)E2 * 8, B), B, 0, stream>>>(ei, as1, ad1, m1, E, E2);
  k_esum1<<<cdiv((long long)E2 * 8, B), B, 0, stream>>>(ei, as1, ad1, m1, s1, E, E2);
  k_agg1<<<cdiv((long long)E2 * 64, B), B, 0, stream>>>(ei, h1, as1, ad1, m1, s1, o1, E, E2);
  k_elubias<<<cdiv((long long)N * 256, B), B, 0, stream>>>(o1, b1, N);
  k_gemm2<<<cdiv((long long)MT * 32, B), B, 0, stream>>>(o1, W2, h2, MT);
  k_att2<<<cdiv(N, B), B, 0, stream>>>(h2, ats2, atd2, as2, ad2, N);
  k_emax2<<<cdiv(E2, B), B, 0, stream>>>(ei, as2, ad2, m2, E, E2);
  k_esum2<<<cdiv(E2, B), B, 0, stream>>>(ei, as2, ad2, m2, s2, E, E2);
  k_agg2<<<cdiv((long long)E2 * 4, B), B, 0, stream>>>(ei, h2, as2, ad2, m2, s2, out, E, E2);
  k_bias2<<<cdiv((long long)N * 16, B), B, 0, stream>>>(out, b2, N);
}